// svd_nufft_48704929136779
// MI455X (gfx1250) — compile-verified
//
#include <hip/hip_runtime.h>
#include <stddef.h>
#include <stdint.h>

typedef float v2f __attribute__((ext_vector_type(2)));
typedef float v8f __attribute__((ext_vector_type(8)));

#define OS 512
#define IM 256
#define NSVD 5
#define NCOIL 8
#define NIMG (NCOIL * NSVD)   // 40 planes (coil-major, svd fastest)
#define NTRJ 262144
#define SG 50                 // temporal grid side

#define TM 32
#define TN 32
#define KC 32
#define PI_F 3.14159265358979323846f

// ---------------------------------------------------------------------------
// CDNA5 async global->LDS copy (bypasses VGPRs, tracked by ASYNCcnt).
// LDS offset = low 32 bits of the generic pointer (aperture tag is [63:32]).
// ---------------------------------------------------------------------------
__device__ __forceinline__ void async_copy_b32(float* lds_dst, const float* gsrc) {
  unsigned lds = (unsigned)(uintptr_t)lds_dst;
  unsigned long long ga = (unsigned long long)(uintptr_t)gsrc;
  asm volatile("global_load_async_to_lds_b32 %0, %1, off"
               :: "v"(lds), "v"(ga) : "memory");
}
__device__ __forceinline__ void wait_async0() {
  asm volatile("s_wait_asynccnt 0x0" ::: "memory");
}

// ---------------------------------------------------------------------------
// Build centered-DFT factor F[k][m] = exp(-2*pi*i*(k-256)*(m-128)/512)/sqrt(512)
// (equivalent to fftshift(fft(ifftshift(pad(x)))) along one axis, ortho norm)
// ---------------------------------------------------------------------------
__global__ void build_dft_kernel(float* __restrict__ Fr, float* __restrict__ Fi) {
  int idx = blockIdx.x * blockDim.x + threadIdx.x;
  if (idx >= OS * IM) return;
  int k = idx / IM;
  int m = idx % IM;
  int p = ((k - OS / 2) * (m - IM / 2)) % OS;   // reduce phase before sincos
  float th = -2.0f * PI_F * (float)p / (float)OS;
  float s, c;
  __sincosf(th, &s, &c);
  const float scale = 0.04419417382415922f;     // 1/sqrt(512)
  Fr[idx] = c * scale;
  Fi[idx] = s * scale;
}

// ---------------------------------------------------------------------------
// X[c,l] = img[c] * spatial[l]  (complex), written as split re/im planes
// ---------------------------------------------------------------------------
__global__ void stage_a_kernel(const float* __restrict__ imr,
                               const float* __restrict__ imi,
                               const float* __restrict__ sf, // (5,256,256) interleaved c64
                               float* __restrict__ Xr, float* __restrict__ Xi) {
  int idx = blockIdx.x * blockDim.x + threadIdx.x;
  if (idx >= NIMG * IM * IM) return;
  int pix = idx % (IM * IM);
  int img = idx / (IM * IM);
  int c = img / NSVD;
  int l = img % NSVD;
  float ar = imr[c * IM * IM + pix];
  float ai = imi[c * IM * IM + pix];
  float sr = sf[((size_t)l * IM * IM + pix) * 2 + 0];
  float si = sf[((size_t)l * IM * IM + pix) * 2 + 1];
  Xr[idx] = ar * sr - ai * si;
  Xi[idx] = ar * si + ai * sr;
}

// ---------------------------------------------------------------------------
// Complex GEMM C = A @ B via 4 real WMMA accumulators (fp32 matrix cores).
// A: row-major M x K (lda), per-image stride aImg (0 => shared).
// B: element (k,n) at Br[k*sbk + n*sbn], per-image stride bImg.
// C: row-major M x N (ldc), per-image stride cImg. grid.z = image.
// Block: 128 threads = 4 waves, 32x32 block tile, K in 32-wide LDS chunks.
// Double-buffered: chunk k+1 streams to LDS via async-to-LDS while chunk k
// runs through the WMMAs; one s_wait_asynccnt + barrier per chunk.
// ---------------------------------------------------------------------------
__global__ __launch_bounds__(128)
void cgemm_wmma_kernel(const float* __restrict__ Ar, const float* __restrict__ Ai,
                       const float* __restrict__ Br, const float* __restrict__ Bi,
                       float* __restrict__ Cr, float* __restrict__ Ci,
                       int K, int lda, int sbk, int sbn, int ldc,
                       long long aImg, long long bImg, long long cImg) {
  __shared__ float Asr[2][TM][KC + 1];
  __shared__ float Asi[2][TM][KC + 1];
  __shared__ float Bsr[2][KC][TN + 1];
  __shared__ float Bsi[2][KC][TN + 1];

  const int tid  = threadIdx.x;
  const int lane = tid & 31;
  const int wav  = tid >> 5;
  const int half = lane >> 4;   // K-pair select for A/B frags, M-half for C/D
  const int lr   = lane & 15;
  const int wm   = (wav >> 1) * 16;  // wave tile row offset in block tile
  const int wn   = (wav & 1) * 16;   // wave tile col offset

  const int rowBase = blockIdx.y * TM;
  const int colBase = blockIdx.x * TN;
  const size_t aOff = (size_t)(aImg * (long long)blockIdx.z);
  const size_t bOff = (size_t)(bImg * (long long)blockIdx.z);
  const size_t cOff = (size_t)(cImg * (long long)blockIdx.z);

  // Issue one K-chunk of async global->LDS copies into buffer `buf`.
  auto stage = [&](int buf, int k0) {
    for (int i = tid; i < TM * KC; i += 128) {
      int r = i / KC, kk = i % KC;           // kk fastest: A is row-major -> coalesced
      size_t g = aOff + (size_t)(rowBase + r) * lda + (size_t)(k0 + kk);
      async_copy_b32(&Asr[buf][r][kk], Ar + g);
      async_copy_b32(&Asi[buf][r][kk], Ai + g);
    }
    for (int i = tid; i < KC * TN; i += 128) {
      int kk, cc;                            // iterate the unit-stride index fastest
      if (sbn == 1) { kk = i / TN; cc = i % TN; }
      else          { cc = i / KC; kk = i % KC; }
      size_t g = bOff + (size_t)(k0 + kk) * sbk + (size_t)(colBase + cc) * sbn;
      async_copy_b32(&Bsr[buf][kk][cc], Br + g);
      async_copy_b32(&Bsi[buf][kk][cc], Bi + g);
    }
  };

  v8f accRR = {};  // Ar*Br
  v8f accII = {};  // Ai*Bi
  v8f accRI = {};  // Ar*Bi
  v8f accIR = {};  // Ai*Br

  int buf = 0;
  stage(0, 0);
  wait_async0();
  __syncthreads();

  for (int k0 = 0; k0 < K; k0 += KC) {
    if (k0 + KC < K) stage(buf ^ 1, k0 + KC);   // prefetch next chunk

#pragma unroll
    for (int ks = 0; ks < KC; ks += 4) {
      const int ka = ks + 2 * half;
      v2f aR, aI, bR, bI;
      // 16x4 A frag: lanes 0-15 K={ks,ks+1}, lanes 16-31 K={ks+2,ks+3}
      aR.x = Asr[buf][wm + lr][ka];     aR.y = Asr[buf][wm + lr][ka + 1];
      aI.x = Asi[buf][wm + lr][ka];     aI.y = Asi[buf][wm + lr][ka + 1];
      // 4x16 B frag mirrors A with rows=K, lanes=N
      bR.x = Bsr[buf][ka][wn + lr];     bR.y = Bsr[buf][ka + 1][wn + lr];
      bI.x = Bsi[buf][ka][wn + lr];     bI.y = Bsi[buf][ka + 1][wn + lr];
      accRR = __builtin_amdgcn_wmma_f32_16x16x4_f32(false, aR, false, bR, (short)0, accRR, false, false);
      accII = __builtin_amdgcn_wmma_f32_16x16x4_f32(false, aI, false, bI, (short)0, accII, false, false);
      accRI = __builtin_amdgcn_wmma_f32_16x16x4_f32(false, aR, false, bI, (short)0, accRI, false, false);
      accIR = __builtin_amdgcn_wmma_f32_16x16x4_f32(false, aI, false, bR, (short)0, accIR, false, false);
    }

    wait_async0();      // next chunk fully resident in LDS
    __syncthreads();    // all waves done reading current chunk
    buf ^= 1;
  }

  // D layout: VGPR j holds (M = j + 8*half, N = lr)
#pragma unroll
  for (int j = 0; j < 8; j++) {
    int row = rowBase + wm + 8 * half + j;
    int col = colBase + wn + lr;
    size_t g = cOff + (size_t)row * ldc + (size_t)col;
    Cr[g] = accRR[j] - accII[j];
    Ci[g] = accRI[j] + accIR[j];
  }
}

// ---------------------------------------------------------------------------
// Per trajectory point: nearest-grid gather from k_os (L2-resident, 84 MB),
// bilinear temporal interp, length-5 complex contraction per coil.
// ---------------------------------------------------------------------------
__global__ void gather_contract_kernel(const float* __restrict__ trj,
                                       const float* __restrict__ tg, // (50,50,5) c64
                                       const float* __restrict__ Kr,
                                       const float* __restrict__ Ki,
                                       float* __restrict__ out) {
  int k = blockIdx.x * blockDim.x + threadIdx.x;
  if (k >= NTRJ) return;
  float t0 = trj[2 * k + 0] * 2.0f;   // OSF
  float t1 = trj[2 * k + 1] * 2.0f;
  float r0 = rintf(t0), r1 = rintf(t1);   // round-half-even == jnp.round
  float d0 = t0 - r0, d1 = t1 - r1;
  int i0 = (((int)r0) + OS / 2) & (OS - 1);
  int i1 = (((int)r1) + OS / 2) & (OS - 1);

  float c0 = fminf(fmaxf((d0 + 0.5f) * (float)(SG - 1), 0.0f), (float)(SG - 1));
  float c1 = fminf(fmaxf((d1 + 0.5f) * (float)(SG - 1), 0.0f), (float)(SG - 1));
  int j0 = (int)floorf(c0); if (j0 > SG - 2) j0 = SG - 2;
  int j1 = (int)floorf(c1); if (j1 > SG - 2) j1 = SG - 2;
  float f0 = c0 - (float)j0, f1 = c1 - (float)j1;
  float w00 = (1.0f - f0) * (1.0f - f1), w01 = (1.0f - f0) * f1;
  float w10 = f0 * (1.0f - f1),          w11 = f0 * f1;

  const float* p00 = tg + ((size_t)j0 * SG + j1) * NSVD * 2;
  const float* p01 = p00 + NSVD * 2;
  const float* p10 = p00 + (size_t)SG * NSVD * 2;
  const float* p11 = p10 + NSVD * 2;

  float tr[NSVD], ti[NSVD];
#pragma unroll
  for (int l = 0; l < NSVD; l++) {
    tr[l] = w00 * p00[2 * l]     + w01 * p01[2 * l]     + w10 * p10[2 * l]     + w11 * p11[2 * l];
    ti[l] = w00 * p00[2 * l + 1] + w01 * p01[2 * l + 1] + w10 * p10[2 * l + 1] + w11 * p11[2 * l + 1];
  }

  size_t addr = (size_t)i0 * OS + (size_t)i1;
#pragma unroll
  for (int c = 0; c < NCOIL; c++) {
    float oR = 0.0f, oI = 0.0f;
#pragma unroll
    for (int l = 0; l < NSVD; l++) {
      size_t pl = (size_t)(c * NSVD + l) * OS * OS + addr;
      float gr = Kr[pl], gi = Ki[pl];
      oR += gr * tr[l] - gi * ti[l];
      oI += gr * ti[l] + gi * tr[l];
    }
    out[((size_t)c * NTRJ + (size_t)k) * 2 + 0] = oR;
    out[((size_t)c * NTRJ + (size_t)k) * 2 + 1] = oI;
  }
}

// ---------------------------------------------------------------------------
extern "C" void kernel_launch(void* const* d_in, const int* in_sizes, int n_in,
                              void* d_out, int out_size, void* d_ws, size_t ws_size,
                              hipStream_t stream) {
  (void)in_sizes; (void)n_in; (void)out_size; (void)ws_size;
  const float* imr = (const float*)d_in[0];   // (1,8,256,256) f32
  const float* imi = (const float*)d_in[1];   // (1,8,256,256) f32
  const float* trj = (const float*)d_in[2];   // (1,262144,2)  f32
  const float* sf  = (const float*)d_in[3];   // (5,256,256)   c64 interleaved
  const float* tg  = (const float*)d_in[4];   // (50,50,5)     c64 interleaved
  float* out = (float*)d_out;                 // (1,8,262144)  c64 interleaved

  float* w = (float*)d_ws;
  size_t o = 0;
  float* Fr = w + o; o += (size_t)OS * IM;
  float* Fi = w + o; o += (size_t)OS * IM;
  float* Xr = w + o; o += (size_t)NIMG * IM * IM;
  float* Xi = w + o; o += (size_t)NIMG * IM * IM;
  float* Yr = w + o; o += (size_t)NIMG * OS * IM;
  float* Yi = w + o; o += (size_t)NIMG * OS * IM;
  float* Kr = w + o; o += (size_t)NIMG * OS * OS;
  float* Ki = w + o; o += (size_t)NIMG * OS * OS;

  build_dft_kernel<<<(OS * IM + 255) / 256, 256, 0, stream>>>(Fr, Fi);
  stage_a_kernel<<<(NIMG * IM * IM + 255) / 256, 256, 0, stream>>>(imr, imi, sf, Xr, Xi);

  // GEMM1: Y(512x256) = F(512x256) @ X(256x256), per image (F shared)
  dim3 g1(IM / TN, OS / TM, NIMG);
  cgemm_wmma_kernel<<<g1, 128, 0, stream>>>(Fr, Fi, Xr, Xi, Yr, Yi,
      /*K=*/IM, /*lda=*/IM, /*sbk=*/IM, /*sbn=*/1, /*ldc=*/IM,
      /*aImg=*/0LL, /*bImg=*/(long long)IM * IM, /*cImg=*/(long long)OS * IM);

  // GEMM2: Kos(512x512) = Y(512x256) @ F^T  (B(k,n) = F[n*IM + k], F shared)
  dim3 g2(OS / TN, OS / TM, NIMG);
  cgemm_wmma_kernel<<<g2, 128, 0, stream>>>(Yr, Yi, Fr, Fi, Kr, Ki,
      /*K=*/IM, /*lda=*/IM, /*sbk=*/1, /*sbn=*/IM, /*ldc=*/OS,
      /*aImg=*/(long long)OS * IM, /*bImg=*/0LL, /*cImg=*/(long long)OS * OS);

  gather_contract_kernel<<<NTRJ / 256, 256, 0, stream>>>(trj, tg, Kr, Ki, out);
}